// STZINBGNN_70153995813106
// MI455X (gfx1250) — compile-verified
//
#include <hip/hip_runtime.h>
#include <math.h>

// ---------------------------------------------------------------------------
// STZINB-GNN forward for MI455X (gfx1250, wave32, WMMA).
// fp32 end-to-end; matrix ops via V_WMMA_F32_16X16X4_F32 (one 16x16 tile/wave).
// ---------------------------------------------------------------------------

typedef __attribute__((ext_vector_type(2))) float v2f;
typedef __attribute__((ext_vector_type(8))) float v8f;

#define TOTN   131072      // B*N nodes
#define NPG_SH 12          // log2(4096) nodes per graph
#define NEDGE  1048576
#define WPB    8           // waves per 256-thread block

// ---------------- small utility kernels ----------------

__global__ void k_fill(float* __restrict__ p, float v, long n) {
  long i = (long)blockIdx.x * blockDim.x + threadIdx.x;
  if (i < n) p[i] = v;
}

__global__ void k_deg_count(const int* __restrict__ dst, float* __restrict__ deg, int ne) {
  int e = blockIdx.x * blockDim.x + threadIdx.x;
  if (e < ne) atomicAdd(&deg[dst[e]], 1.0f);
}

__global__ void k_rsqrt(float* __restrict__ p, int n) {
  int i = blockIdx.x * blockDim.x + threadIdx.x;
  if (i < n) p[i] = rsqrtf(p[i]);
}

// ---------------- dense GEMM: Y[M,N] = X[M,K] @ W[K,N] (fp32 WMMA) ----------

__global__ void k_gemm_wmma(const float* __restrict__ X, const float* __restrict__ W,
                            float* __restrict__ Y, int K, int Nout) {
  const int lane = threadIdx.x & 31;
  const int wave = threadIdx.x >> 5;
  const int Nt   = Nout >> 4;
  const long tile = (long)blockIdx.x * WPB + wave;     // grid sized exactly
  const int tm = (int)(tile / Nt);
  const int tn = (int)(tile % Nt);
  const int mrow = tm * 16 + (lane & 15);              // A-frag row for this lane
  const int ncol = tn * 16 + (lane & 15);              // B/C-frag column
  const int ksub = (lane >> 4) << 1;                   // lanes 16-31 hold K+2,K+3

  v8f acc = {};
  for (int k = 0; k < K; k += 4) {
    v2f a, b;
    const float* xp = X + (size_t)mrow * K + (k + ksub);
    a.x = xp[0];
    a.y = xp[1];
    const float* wp = W + (size_t)(k + ksub) * Nout + ncol;
    b.x = wp[0];
    b.y = wp[Nout];
    acc = __builtin_amdgcn_wmma_f32_16x16x4_f32(false, a, false, b,
                                                (short)0, acc, false, false);
  }
  // C/D layout: VGPR r -> (M = r + 8*(lane>=16), N = lane&15)
  const int mbase = tm * 16 + ((lane >> 4) << 3);
  float* yp = Y + (size_t)mbase * Nout + ncol;
#pragma unroll
  for (int r = 0; r < 8; ++r) yp[(size_t)r * Nout] = acc[r];
}

// ---------------- GCN edge scatter: agg[dst] += h[src] * dinv[s]*dinv[d] ----

__global__ void k_scatter(const float* __restrict__ H, const int* __restrict__ src,
                          const int* __restrict__ dst, const float* __restrict__ dinv,
                          float* __restrict__ agg, int C) {
  const int per = C >> 2;                              // threads per edge (float4 each)
  long tid = (long)blockIdx.x * blockDim.x + threadIdx.x;
  long e = tid / per;
  int  f = (int)(tid % per) << 2;
  if (e >= NEDGE) return;
  const int s = src[e], d = dst[e];
  const float w = dinv[s] * dinv[d];
  const float4 hv = *(const float4*)(H + (size_t)s * C + f);
  float* ap = agg + (size_t)d * C + f;
  atomicAdd(ap + 0, hv.x * w);
  atomicAdd(ap + 1, hv.y * w);
  atomicAdd(ap + 2, hv.z * w);
  atomicAdd(ap + 3, hv.w * w);
}

// ---------------- GCN finalize: Y = relu(agg + H*dinv^2 + b), in place ------

__global__ void k_gcn_fin(float* __restrict__ agg, const float* __restrict__ H,
                          const float* __restrict__ dinv, const float* __restrict__ bias,
                          int C) {
  long i = (long)blockIdx.x * blockDim.x + threadIdx.x;   // grid exact: TOTN*C
  long node = i / C;
  int  f = (int)(i % C);
  float sn = dinv[node];
  sn *= sn;
  float v = agg[i] + H[i] * sn + bias[f];
  agg[i] = fmaxf(v, 0.0f);
}

// ---------------- conv1d(kernel=3,pad=1) as 3-tap WMMA GEMM -----------------
// y[node,o] = relu( b[o] + sum_t sum_c TW[o,c,t] * X[node+t-1, c] )
// graph boundaries are 4096-aligned -> validity test (g>>12)==(m>>12).

__global__ void k_conv_wmma(const float* __restrict__ X, const float* __restrict__ TW,
                            const float* __restrict__ bias, float* __restrict__ Y,
                            int K, int Nout) {
  const int lane = threadIdx.x & 31;
  const int wave = threadIdx.x >> 5;
  const int Nt   = Nout >> 4;
  const long tile = (long)blockIdx.x * WPB + wave;
  const int tm = (int)(tile / Nt);
  const int tn = (int)(tile % Nt);
  const int mrow = tm * 16 + (lane & 15);
  const int ncol = tn * 16 + (lane & 15);
  const int ksub = (lane >> 4) << 1;

  v8f acc = {};
#pragma unroll
  for (int t = 0; t < 3; ++t) {
    int g = mrow + t - 1;
    const float sel = ((g >> NPG_SH) == (mrow >> NPG_SH)) ? 1.0f : 0.0f;
    g = g < 0 ? 0 : (g >= TOTN ? TOTN - 1 : g);        // clamp so loads stay in-bounds
    const float* xrow = X + (size_t)g * K;
    const float* wcol = TW + (size_t)ncol * K * 3 + t; // TW[o][c][t], o=ncol
    for (int k = 0; k < K; k += 4) {
      v2f a, b;
      a.x = xrow[k + ksub] * sel;
      a.y = xrow[k + ksub + 1] * sel;
      b.x = wcol[(size_t)(k + ksub) * 3];
      b.y = wcol[(size_t)(k + ksub + 1) * 3];
      acc = __builtin_amdgcn_wmma_f32_16x16x4_f32(false, a, false, b,
                                                  (short)0, acc, false, false);
    }
  }
  const int mbase = tm * 16 + ((lane >> 4) << 3);
  const float bn = bias[ncol];
  float* yp = Y + (size_t)mbase * Nout + ncol;
#pragma unroll
  for (int r = 0; r < 8; ++r) yp[(size_t)r * Nout] = fmaxf(acc[r] + bn, 0.0f);
}

// ---------------- heads: wave-per-node dot products + activations -----------

__global__ void k_heads(const float* __restrict__ XT,
                        const float* __restrict__ Wpi, const float* __restrict__ bpi,
                        const float* __restrict__ Wn,  const float* __restrict__ bn,
                        const float* __restrict__ Wp,  const float* __restrict__ bp,
                        float* __restrict__ out, int C) {
  const int lane = threadIdx.x & 31;
  const long node = (long)blockIdx.x * (blockDim.x >> 5) + (threadIdx.x >> 5);
  float s0 = 0.f, s1 = 0.f, s2 = 0.f;
  for (int f = lane * 4; f < C; f += 128) {
    const float4 xv = *(const float4*)(XT + (size_t)node * C + f);
    const float4 w0 = *(const float4*)(Wpi + f);
    const float4 w1 = *(const float4*)(Wn + f);
    const float4 w2 = *(const float4*)(Wp + f);
    s0 += xv.x * w0.x + xv.y * w0.y + xv.z * w0.z + xv.w * w0.w;
    s1 += xv.x * w1.x + xv.y * w1.y + xv.z * w1.z + xv.w * w1.w;
    s2 += xv.x * w2.x + xv.y * w2.y + xv.z * w2.z + xv.w * w2.w;
  }
#pragma unroll
  for (int off = 16; off > 0; off >>= 1) {
    s0 += __shfl_down(s0, off, 32);
    s1 += __shfl_down(s1, off, 32);
    s2 += __shfl_down(s2, off, 32);
  }
  if (lane == 0) {
    const float a = s0 + bpi[0];
    const float b = s1 + bn[0];
    const float c = s2 + bp[0];
    out[node]             = 1.0f / (1.0f + expf(-a));
    out[(long)TOTN + node]  = fmaxf(b, 0.0f) + log1pf(expf(-fabsf(b)));  // softplus
    out[2L * TOTN + node] = 1.0f / (1.0f + expf(-c));
  }
}

// ---------------------------------------------------------------------------

extern "C" void kernel_launch(void* const* d_in, const int* in_sizes, int n_in,
                              void* d_out, int out_size, void* d_ws, size_t ws_size,
                              hipStream_t stream) {
  const float* x   = (const float*)d_in[0];
  const int*   ei  = (const int*)d_in[1];       // [2, E] (jax int64 -> int32 w/o x64)
  const int* srcI  = ei;
  const int* dstI  = ei + NEDGE;
  const float* W1  = (const float*)d_in[3];
  const float* b1  = (const float*)d_in[4];
  const float* W2  = (const float*)d_in[5];
  const float* b2  = (const float*)d_in[6];
  const float* W3  = (const float*)d_in[7];
  const float* b3  = (const float*)d_in[8];
  const float* tW1 = (const float*)d_in[9];
  const float* tb1 = (const float*)d_in[10];
  const float* tW2 = (const float*)d_in[11];
  const float* tb2 = (const float*)d_in[12];
  const float* tW3 = (const float*)d_in[13];
  const float* tb3 = (const float*)d_in[14];
  const float* Wpi = (const float*)d_in[15];
  const float* bpi = (const float*)d_in[16];
  const float* Wn  = (const float*)d_in[17];
  const float* bn  = (const float*)d_in[18];
  const float* Wp  = (const float*)d_in[19];
  const float* bp  = (const float*)d_in[20];

  float* ws   = (float*)d_ws;
  float* dinv = ws;                              // TOTN
  float* bufA = ws + TOTN;                       // TOTN*128
  float* bufB = bufA + (size_t)TOTN * 128;       // TOTN*128
  float* out  = (float*)d_out;

  const int T = 256;

  // --- degree / normalization -------------------------------------------
  k_fill<<<TOTN / T, T, 0, stream>>>(dinv, 1.0f, (long)TOTN);       // self-loop
  k_deg_count<<<NEDGE / T, T, 0, stream>>>(dstI, dinv, NEDGE);
  k_rsqrt<<<TOTN / T, T, 0, stream>>>(dinv, TOTN);

  // --- GCN layer: H=X@W; agg=scatter; Y=relu(agg + H*snorm + b) (into bufB)
  auto gcn_layer = [&](const float* Xin, int K, int C, const float* W,
                       const float* bias) {
    const long tiles = (long)(TOTN / 16) * (C / 16);
    k_gemm_wmma<<<(int)(tiles / WPB), T, 0, stream>>>(Xin, W, bufA, K, C);
    k_fill<<<(int)((long)TOTN * C / T), T, 0, stream>>>(bufB, 0.0f, (long)TOTN * C);
    const long sth = (long)NEDGE * (C / 4);
    k_scatter<<<(int)(sth / T), T, 0, stream>>>(bufA, srcI, dstI, dinv, bufB, C);
    k_gcn_fin<<<(int)((long)TOTN * C / T), T, 0, stream>>>(bufB, bufA, dinv, bias, C);
  };

  gcn_layer(x,    64, 128, W1, b1);   // -> bufB [TOTN,128]
  gcn_layer(bufB, 128, 32, W2, b2);   // -> bufB [TOTN,32]
  gcn_layer(bufB, 32, 128, W3, b3);   // -> bufB [TOTN,128]

  // --- temporal conv stack ----------------------------------------------
  auto conv_layer = [&](const float* Xin, float* Yout, int K, int C,
                        const float* TW, const float* bias) {
    const long tiles = (long)(TOTN / 16) * (C / 16);
    k_conv_wmma<<<(int)(tiles / WPB), T, 0, stream>>>(Xin, TW, bias, Yout, K, C);
  };

  conv_layer(bufB, bufA, 128, 128, tW1, tb1);
  conv_layer(bufA, bufB, 128, 32,  tW2, tb2);
  conv_layer(bufB, bufA, 32,  128, tW3, tb3);

  // --- output heads ------------------------------------------------------
  k_heads<<<TOTN / (T / 32), T, 0, stream>>>(bufA, Wpi, bpi, Wn, bn, Wp, bp, out, 128);
}